// SetAbstraction_22531398435385
// MI455X (gfx1250) — compile-verified
//
#include <hip/hip_runtime.h>
#include <hip/hip_bf16.h>
#include <math.h>

typedef __attribute__((ext_vector_type(2))) float v2f;
typedef __attribute__((ext_vector_type(8))) float v8f;

#define NPTS 4096
#define NPOINT 1024
#define NSAMP 32
#define CF 64
#define C0P 68          // 3 + 64 = 67, padded to 68 for k%4==0
#define BN_EPS 1e-3f

// ---------------------------------------------------------------------------
// Kernel 1: farthest point sampling. One block per batch, 1024 threads.
// dist[] and the xyz cloud live in LDS; block-wide argmax per step with
// lowest-index tie-break (matches jnp.argmax).
// ---------------------------------------------------------------------------
__global__ void __launch_bounds__(1024) fps_kernel(const float* __restrict__ xyz,
                                                   float* __restrict__ out,      // d_out (new_xyz at offset 0)
                                                   float* __restrict__ nxyz_ws)  // ws copy of new_xyz
{
    extern __shared__ float smem[];
    float* px   = smem;            // 4096
    float* py   = px + NPTS;       // 4096
    float* pz   = py + NPTS;       // 4096
    float* dist = pz + NPTS;       // 4096
    float* rd   = dist + NPTS;     // 1024
    int*   ri   = (int*)(rd + 1024); // 1024
    __shared__ int s_far;

    const int b = blockIdx.x;
    const int t = threadIdx.x;
    const float* X = xyz + (size_t)b * NPTS * 3;

    for (int i = t; i < NPTS; i += 1024) {
        px[i] = X[i * 3 + 0];
        py[i] = X[i * 3 + 1];
        pz[i] = X[i * 3 + 2];
        dist[i] = 1e10f;
    }
    if (t == 0) s_far = 0;
    __syncthreads();

    for (int it = 0; it < NPOINT; ++it) {
        const int far = s_far;
        const float cx = px[far], cy = py[far], cz = pz[far];
        float bd = -1.0f; int bi = 0x7fffffff;
        for (int i = t; i < NPTS; i += 1024) {
            float dx = px[i] - cx, dy = py[i] - cy, dz = pz[i] - cz;
            float d2 = dx * dx + dy * dy + dz * dz;
            float nd = fminf(dist[i], d2);
            dist[i] = nd;
            if (nd > bd) { bd = nd; bi = i; }   // ascending i within thread
        }
        rd[t] = bd; ri[t] = bi;
        __syncthreads();
        for (int off = 512; off > 0; off >>= 1) {
            if (t < off) {
                float od = rd[t + off]; int oi = ri[t + off];
                if (od > rd[t] || (od == rd[t] && oi < ri[t])) { rd[t] = od; ri[t] = oi; }
            }
            __syncthreads();
        }
        if (t == 0) {
            int nf = ri[0];
            s_far = nf;
            float vx = px[nf], vy = py[nf], vz = pz[nf];
            size_t o = ((size_t)b * NPOINT + it) * 3;
            out[o] = vx; out[o + 1] = vy; out[o + 2] = vz;
            nxyz_ws[o] = vx; nxyz_ws[o + 1] = vy; nxyz_ws[o + 2] = vz;
        }
        __syncthreads();
    }
}

// ---------------------------------------------------------------------------
// Kernel 2: 32-NN for each centroid. One thread per centroid; xyz tiled
// through LDS; insertion-sorted top-32 kept ascending (stable on ties).
// ---------------------------------------------------------------------------
#define KNN_T 128
#define KTILE 1024
__global__ void __launch_bounds__(KNN_T) knn_kernel(const float* __restrict__ xyz,
                                                    const float* __restrict__ nxyz,
                                                    int* __restrict__ knn)
{
    __shared__ float tx[KTILE], ty[KTILE], tz[KTILE];
    const int g = blockIdx.x * KNN_T + threadIdx.x;   // centroid id, [0, 16384)
    const int b = g >> 10;
    const float* X = xyz + (size_t)b * NPTS * 3;
    const float cx = nxyz[g * 3 + 0], cy = nxyz[g * 3 + 1], cz = nxyz[g * 3 + 2];

    float bd[NSAMP]; int bi[NSAMP];
#pragma unroll
    for (int k = 0; k < NSAMP; ++k) { bd[k] = 3.4e38f; bi[k] = 0; }

    for (int t0 = 0; t0 < NPTS; t0 += KTILE) {
        __syncthreads();
        for (int i = threadIdx.x; i < KTILE; i += KNN_T) {
            tx[i] = X[(t0 + i) * 3 + 0];
            ty[i] = X[(t0 + i) * 3 + 1];
            tz[i] = X[(t0 + i) * 3 + 2];
        }
        __syncthreads();
        for (int i = 0; i < KTILE; ++i) {
            float dx = tx[i] - cx, dy = ty[i] - cy, dz = tz[i] - cz;
            float d2 = fmaxf(dx * dx + dy * dy + dz * dz, 0.0f);
            if (d2 < bd[NSAMP - 1]) {
                int p = NSAMP - 1;
                while (p > 0 && bd[p - 1] > d2) { bd[p] = bd[p - 1]; bi[p] = bi[p - 1]; --p; }
                bd[p] = d2; bi[p] = t0 + i;
            }
        }
    }
    for (int k = 0; k < NSAMP; ++k) knn[(size_t)g * NSAMP + k] = bi[k];
}

// ---------------------------------------------------------------------------
// Kernel 3: fused gather + 3-layer 1x1-conv MLP (WMMA f32 16x16x4) + max-pool.
// 4 waves per block, one centroid group (32x67 tile) per wave. Weights + BN
// params resident in LDS. BN folded: y = relu(z + b) * s + t.
// ---------------------------------------------------------------------------
#define MLP_T 128
#define GPB 4

__device__ __forceinline__ void gemm_layer(const float* __restrict__ IN, int inStride, int Kdim,
                                           const float* __restrict__ W, int N,
                                           const float* __restrict__ bias,
                                           const float* __restrict__ sc,
                                           const float* __restrict__ tr,
                                           float* __restrict__ OUT, int outStride,
                                           int rrow, int half)
{
    for (int mt = 0; mt < 32; mt += 16) {
        for (int nt = 0; nt < N; nt += 16) {
            v8f acc = {};
            for (int kk = 0; kk < Kdim; kk += 4) {
                v2f a, bb;
                const float* ap = IN + (mt + rrow) * inStride + kk + 2 * half;
                a.x = ap[0]; a.y = ap[1];
                const float* bp = W + (kk + 2 * half) * N + nt + rrow;
                bb.x = bp[0]; bb.y = bp[N];
                acc = __builtin_amdgcn_wmma_f32_16x16x4_f32(false, a, false, bb,
                                                            (short)0, acc, false, false);
            }
            const int col = nt + rrow;
            const float sb = bias[col], ss = sc[col], tt = tr[col];
#pragma unroll
            for (int j = 0; j < 8; ++j) {
                float v = acc[j] + sb;
                v = fmaxf(v, 0.0f);
                v = v * ss + tt;
                OUT[(mt + j + 8 * half) * outStride + col] = v;
            }
        }
    }
}

__global__ void __launch_bounds__(MLP_T) samlp_kernel(
    const float* __restrict__ xyz, const float* __restrict__ feat,
    const float* __restrict__ nxyz, const int* __restrict__ knn,
    const float* __restrict__ W0, const float* __restrict__ b0, const float* __restrict__ g0,
    const float* __restrict__ be0, const float* __restrict__ m0, const float* __restrict__ v0,
    const float* __restrict__ W1, const float* __restrict__ b1, const float* __restrict__ g1,
    const float* __restrict__ be1, const float* __restrict__ m1, const float* __restrict__ v1,
    const float* __restrict__ W2, const float* __restrict__ b2, const float* __restrict__ g2,
    const float* __restrict__ be2, const float* __restrict__ m2, const float* __restrict__ v2,
    float* __restrict__ out)
{
    extern __shared__ float lds[];
    float* w0  = lds;                 // 68*64
    float* w1  = w0 + C0P * 64;       // 64*64
    float* w2  = w1 + 64 * 64;        // 64*128
    float* pb0 = w2 + 64 * 128;  float* ps0 = pb0 + 64;  float* pt0 = ps0 + 64;
    float* pb1 = pt0 + 64;       float* ps1 = pb1 + 64;  float* pt1 = ps1 + 64;
    float* pb2 = pt1 + 64;       float* ps2 = pb2 + 128; float* pt2 = ps2 + 128;
    float* bufA = pt2 + 128;          // GPB * 32 * 68
    float* bufB = bufA + GPB * 32 * C0P; // GPB * 32 * 64

    const int tid = threadIdx.x;

    // stage weights (pad W0 to 68 rows with zeros) and folded BN params
    for (int i = tid; i < C0P * 64; i += MLP_T) {
        int r = i >> 6, c = i & 63;
        w0[i] = (r < 67) ? W0[r * 64 + c] : 0.0f;
    }
    for (int i = tid; i < 64 * 64;  i += MLP_T) w1[i] = W1[i];
    for (int i = tid; i < 64 * 128; i += MLP_T) w2[i] = W2[i];
    if (tid < 64) {
        float s = g0[tid] * rsqrtf(v0[tid] + BN_EPS);
        ps0[tid] = s; pt0[tid] = be0[tid] - m0[tid] * s; pb0[tid] = b0[tid];
        s = g1[tid] * rsqrtf(v1[tid] + BN_EPS);
        ps1[tid] = s; pt1[tid] = be1[tid] - m1[tid] * s; pb1[tid] = b1[tid];
    }
    if (tid < 128) {
        float s = g2[tid] * rsqrtf(v2[tid] + BN_EPS);
        ps2[tid] = s; pt2[tid] = be2[tid] - m2[tid] * s; pb2[tid] = b2[tid];
    }
    __syncthreads();

    const int wave = tid >> 5, lane = tid & 31;
    const int g = blockIdx.x * GPB + wave;          // group/centroid id
    const int b = g >> 10;
    float* X = bufA + wave * (32 * C0P);            // 32 x 68
    float* Y = bufB + wave * (32 * 64);             // 32 x 64

    // gather: one neighbor row per lane
    {
        int pidx = knn[(size_t)g * NSAMP + lane];
        float cx = nxyz[g * 3 + 0], cy = nxyz[g * 3 + 1], cz = nxyz[g * 3 + 2];
        const float* P = xyz + ((size_t)b * NPTS + pidx) * 3;
        float* row = X + lane * C0P;
        row[0] = P[0] - cx; row[1] = P[1] - cy; row[2] = P[2] - cz;
        const float* F = feat + ((size_t)b * NPTS + pidx) * CF;
        for (int c = 0; c < CF; c += 4) {
            row[3 + c] = F[c];     row[4 + c] = F[c + 1];
            row[5 + c] = F[c + 2]; row[6 + c] = F[c + 3];
        }
        row[67] = 0.0f;
    }
    __syncthreads();

    const int rrow = lane & 15, half = lane >> 4;

    // layer 1: X(32x68) @ w0(68x64) -> Y(32x64)
    gemm_layer(X, C0P, C0P, w0, 64, pb0, ps0, pt0, Y, 64, rrow, half);
    __syncthreads();
    // layer 2: Y(32x64) @ w1(64x64) -> X region (stride 68)
    gemm_layer(Y, 64, 64, w1, 64, pb1, ps1, pt1, X, C0P, rrow, half);
    __syncthreads();

    // layer 3 + max-pool over K=32: X(32x64, stride 68) @ w2(64x128)
    for (int nt = 0; nt < 128; nt += 16) {
        v8f a0 = {}, a1 = {};
        for (int kk = 0; kk < 64; kk += 4) {
            v2f bv;
            const float* bp = w2 + (kk + 2 * half) * 128 + nt + rrow;
            bv.x = bp[0]; bv.y = bp[128];
            v2f aa;
            const float* ap0 = X + (0 + rrow) * C0P + kk + 2 * half;
            aa.x = ap0[0]; aa.y = ap0[1];
            a0 = __builtin_amdgcn_wmma_f32_16x16x4_f32(false, aa, false, bv, (short)0, a0, false, false);
            const float* ap1 = X + (16 + rrow) * C0P + kk + 2 * half;
            aa.x = ap1[0]; aa.y = ap1[1];
            a1 = __builtin_amdgcn_wmma_f32_16x16x4_f32(false, aa, false, bv, (short)0, a1, false, false);
        }
        const int col = nt + rrow;
        const float sb = pb2[col], ss = ps2[col], tt = pt2[col];
        float pm = -3.4e38f;
#pragma unroll
        for (int j = 0; j < 8; ++j) {
            float v = fmaxf(a0[j] + sb, 0.0f) * ss + tt;
            pm = fmaxf(pm, v);
            v = fmaxf(a1[j] + sb, 0.0f) * ss + tt;
            pm = fmaxf(pm, v);
        }
        float other = __shfl_xor(pm, 16, 32);   // combine the two lane halves
        pm = fmaxf(pm, other);
        if (half == 0)
            out[(size_t)16 * NPOINT * 3 + (size_t)g * 128 + col] = pm;
    }
}

// ---------------------------------------------------------------------------
extern "C" void kernel_launch(void* const* d_in, const int* in_sizes, int n_in,
                              void* d_out, int out_size, void* d_ws, size_t ws_size,
                              hipStream_t stream) {
    const float* xyz  = (const float*)d_in[0];
    const float* feat = (const float*)d_in[1];
    const float* W0 = (const float*)d_in[2];  const float* b0 = (const float*)d_in[3];
    const float* g0 = (const float*)d_in[4];  const float* be0 = (const float*)d_in[5];
    const float* m0 = (const float*)d_in[6];  const float* v0 = (const float*)d_in[7];
    const float* W1 = (const float*)d_in[8];  const float* b1 = (const float*)d_in[9];
    const float* g1 = (const float*)d_in[10]; const float* be1 = (const float*)d_in[11];
    const float* m1 = (const float*)d_in[12]; const float* v1 = (const float*)d_in[13];
    const float* W2 = (const float*)d_in[14]; const float* b2 = (const float*)d_in[15];
    const float* g2 = (const float*)d_in[16]; const float* be2 = (const float*)d_in[17];
    const float* m2 = (const float*)d_in[18]; const float* v2 = (const float*)d_in[19];
    float* out = (float*)d_out;

    char* ws = (char*)d_ws;
    float* nxyz = (float*)ws;                        // 16*1024*3 floats = 196608 B
    int*   knn  = (int*)(ws + 196608);               // 16*1024*32 ints = 2 MB

    // 1) farthest point sampling (writes new_xyz into d_out and ws)
    size_t fps_lds = (4 * NPTS + 1024) * sizeof(float) + 1024 * sizeof(int);
    fps_kernel<<<16, 1024, fps_lds, stream>>>(xyz, out, nxyz);

    // 2) 32-NN per centroid
    knn_kernel<<<(16 * NPOINT) / KNN_T, KNN_T, 0, stream>>>(xyz, nxyz, knn);

    // 3) fused gather + MLP (WMMA fp32) + max-pool
    size_t mlp_lds = (size_t)(C0P * 64 + 64 * 64 + 64 * 128 + 3 * 64 + 3 * 64 + 3 * 128
                              + GPB * 32 * C0P + GPB * 32 * 64) * sizeof(float);
    samlp_kernel<<<(16 * NPOINT) / GPB, MLP_T, mlp_lds, stream>>>(
        xyz, feat, nxyz, knn,
        W0, b0, g0, be0, m0, v0,
        W1, b1, g1, be1, m1, v1,
        W2, b2, g2, be2, m2, v2,
        out);
}